// GAT_54958401519775
// MI455X (gfx1250) — compile-verified
//
#include <hip/hip_runtime.h>

typedef __attribute__((ext_vector_type(2))) float v2f;
typedef __attribute__((ext_vector_type(8))) float v8f;

#define N_NODES 50000
#define E_EDGES 400000
#define ETOT    (E_EDGES + N_NODES)   // edges + self loops

// ---------------------------------------------------------------------------
// fp32 WMMA GEMM:  C[M x Nc] = A[M x K] @ B[K x Nc]
// One wave computes a 16x64 macro-tile (4 x 16x16 WMMA accumulators) so each
// A fragment feeds 4 x V_WMMA_F32_16X16X4_F32 and address math is amortized.
// M % 16 == 0, K % 4 == 0, Nc % 64 == 0 guaranteed by the problem sizes.
// ---------------------------------------------------------------------------
__global__ void gemm_wmma_f32(const float* __restrict__ A,
                              const float* __restrict__ B,
                              float* __restrict__ C,
                              int M, int K, int Nc) {
  const int lane = threadIdx.x & 31;
  const int wave = threadIdx.x >> 5;
  const int wavesPerBlock = blockDim.x >> 5;
  const int tilesN = Nc >> 6;                 // 64-wide macro tiles
  const int tilesM = M >> 4;
  const int w = blockIdx.x * wavesPerBlock + wave;
  if (w >= tilesM * tilesN) return;           // uniform per wave
  const int tn = w % tilesN;
  const int tm = w / tilesN;

  const int half = lane >> 4;                 // 0: K=0,1 / M rows 0..7 ; 1: K=2,3 / rows 8..15
  const int r    = lane & 15;
  const int koff = half * 2;

  const float* Ap = A + (size_t)(tm * 16 + r) * K + koff;    // this lane's K pair
  const float* Bp = B + (size_t)koff * Nc + tn * 64 + r;     // 4 columns: r, r+16, r+32, r+48

  v8f acc0 = {}, acc1 = {}, acc2 = {}, acc3 = {};
  for (int k = 0; k < K; k += 4) {
    __builtin_prefetch(Ap + 16, 0, 3);        // global_prefetch_b8, next A chunk
    v2f av;
    av.x = Ap[0];
    av.y = Ap[1];
    v2f b0, b1, b2, b3;
    b0.x = Bp[0];       b0.y = Bp[Nc + 0];
    b1.x = Bp[16];      b1.y = Bp[Nc + 16];
    b2.x = Bp[32];      b2.y = Bp[Nc + 32];
    b3.x = Bp[48];      b3.y = Bp[Nc + 48];
    acc0 = __builtin_amdgcn_wmma_f32_16x16x4_f32(false, av, false, b0, (short)0, acc0, false, false);
    acc1 = __builtin_amdgcn_wmma_f32_16x16x4_f32(false, av, false, b1, (short)0, acc1, false, false);
    acc2 = __builtin_amdgcn_wmma_f32_16x16x4_f32(false, av, false, b2, (short)0, acc2, false, false);
    acc3 = __builtin_amdgcn_wmma_f32_16x16x4_f32(false, av, false, b3, (short)0, acc3, false, false);
    Ap += 4;
    Bp += (size_t)4 * Nc;
  }

  // D layout: lanes 0-15 hold N=lane, M=vgpr; lanes 16-31 hold M=vgpr+8
  float* Cp = C + (size_t)(tm * 16 + half * 8) * Nc + tn * 64 + r;
  #pragma unroll
  for (int i = 0; i < 8; ++i) {
    Cp[(size_t)i * Nc + 0]  = acc0[i];
    Cp[(size_t)i * Nc + 16] = acc1[i];
    Cp[(size_t)i * Nc + 32] = acc2[i];
    Cp[(size_t)i * Nc + 48] = acc3[i];
  }
}

// ---------------------------------------------------------------------------
// Attention helpers
// ---------------------------------------------------------------------------
__device__ __forceinline__ float leaky02(float e) { return e > 0.f ? e : 0.2f * e; }
__device__ __forceinline__ float elu1(float x)    { return x > 0.f ? x : (expf(x) - 1.f); }

__device__ __forceinline__ void atomicMaxF(float* addr, float v) {
  if (v >= 0.f) atomicMax((int*)addr, __float_as_int(v));
  else          atomicMin((unsigned int*)addr, __float_as_uint(v));
}

// ls[n,h] = <lin[n,h,:], a_src[h,:]>, ld likewise
__global__ void node_logits(const float* __restrict__ lin,
                            const float* __restrict__ a_src,
                            const float* __restrict__ a_dst,
                            float* __restrict__ ls, float* __restrict__ ld,
                            int n, int H, int Cc) {
  int t = blockIdx.x * blockDim.x + threadIdx.x;
  if (t >= n * H) return;
  int node = t / H, hh = t % H;
  const float* hp = lin + ((size_t)node * H + hh) * Cc;
  const float* as = a_src + (size_t)hh * Cc;
  const float* ad = a_dst + (size_t)hh * Cc;
  float s = 0.f, d = 0.f;
  for (int c = 0; c < Cc; ++c) { float v = hp[c]; s += v * as[c]; d += v * ad[c]; }
  ls[t] = s; ld[t] = d;
}

__global__ void init_md(float* __restrict__ m, float* __restrict__ denom, int n) {
  int t = blockIdx.x * blockDim.x + threadIdx.x;
  if (t < n) { m[t] = -__builtin_inff(); denom[t] = 0.f; }
}

__global__ void zero_buf(float* __restrict__ p, size_t n) {
  size_t t = (size_t)blockIdx.x * blockDim.x + threadIdx.x;
  if (t < n) p[t] = 0.f;
}

__device__ __forceinline__ void edge_endpoints(const int* __restrict__ ei, int e,
                                               int& s, int& d) {
  if (e < E_EDGES) { s = ei[e]; d = ei[E_EDGES + e]; }
  else             { s = d = e - E_EDGES; }          // self loop
}

__global__ void edge_max(const int* __restrict__ ei,
                         const float* __restrict__ ls, const float* __restrict__ ld,
                         float* __restrict__ m, int H) {
  int t = blockIdx.x * blockDim.x + threadIdx.x;
  if (t >= ETOT * H) return;
  int e = t / H, hh = t % H;
  int s, d; edge_endpoints(ei, e, s, d);
  float val = leaky02(ls[s * H + hh] + ld[d * H + hh]);
  atomicMaxF(&m[d * H + hh], val);
}

__global__ void edge_expsum(const int* __restrict__ ei,
                            const float* __restrict__ ls, const float* __restrict__ ld,
                            const float* __restrict__ m, float* __restrict__ denom,
                            float* __restrict__ ew, int H) {
  int t = blockIdx.x * blockDim.x + threadIdx.x;
  if (t >= ETOT * H) return;
  int e = t / H, hh = t % H;
  int s, d; edge_endpoints(ei, e, s, d);
  float val = leaky02(ls[s * H + hh] + ld[d * H + hh]);
  float w = expf(val - m[d * H + hh]);
  ew[t] = w;
  atomicAdd(&denom[d * H + hh], w);
}

// agg[dst, h, c] += lin[src, h, c] * alpha(e, h)
__global__ void edge_scatter(const int* __restrict__ ei,
                             const float* __restrict__ lin,
                             const float* __restrict__ ew,
                             const float* __restrict__ denom,
                             float* __restrict__ agg, int H, int Cc) {
  int e = blockIdx.x;
  int s, d; edge_endpoints(ei, e, s, d);
  const int HCl = H * Cc;
  for (int j = threadIdx.x; j < HCl; j += blockDim.x) {
    int hh = j / Cc;
    float alpha = ew[(size_t)e * H + hh] / (denom[d * H + hh] + 1e-16f);
    atomicAdd(&agg[(size_t)d * HCl + j], lin[(size_t)s * HCl + j] * alpha);
  }
}

// layers 1-2 (concat): out = elu(agg + b + sb)   (agg already holds skip + attn; in-place)
__global__ void finalize_concat(const float* agg,
                                const float* __restrict__ b,
                                const float* __restrict__ sb,
                                float* out, int n, int HCl) {
  int t = blockIdx.x * blockDim.x + threadIdx.x;
  if (t >= n * HCl) return;
  int j = t % HCl;
  out[t] = elu1(agg[t] + b[j] + sb[j]);
}

// layer 3 (mean over 6 heads): out = elu(mean_h(agg3) + b3 + s3b + skip) ; skip pre-stored in out
__global__ void finalize_mean(const float* __restrict__ agg3,
                              const float* __restrict__ b3,
                              const float* __restrict__ s3b,
                              float* out, int n) {
  int t = blockIdx.x * blockDim.x + threadIdx.x;
  if (t >= n * 128) return;
  int node = t >> 7, j = t & 127;
  float s = 0.f;
  #pragma unroll
  for (int hh = 0; hh < 6; ++hh) s += agg3[(size_t)node * 768 + hh * 128 + j];
  s = s * (1.0f / 6.0f) + b3[j] + s3b[j] + out[t];
  out[t] = elu1(s);
}

// ---------------------------------------------------------------------------
static inline int ceil_div(int a, int b) { return (a + b - 1) / b; }

extern "C" void kernel_launch(void* const* d_in, const int* in_sizes, int n_in,
                              void* d_out, int out_size, void* d_ws, size_t ws_size,
                              hipStream_t stream) {
  const float* x   = (const float*)d_in[0];
  const int*   ei  = (const int*)  d_in[1];
  const float* W1  = (const float*)d_in[2];
  const float* as1 = (const float*)d_in[3];
  const float* ad1 = (const float*)d_in[4];
  const float* b1  = (const float*)d_in[5];
  const float* s1W = (const float*)d_in[6];
  const float* s1b = (const float*)d_in[7];
  const float* W2  = (const float*)d_in[8];
  const float* as2 = (const float*)d_in[9];
  const float* ad2 = (const float*)d_in[10];
  const float* b2  = (const float*)d_in[11];
  const float* s2W = (const float*)d_in[12];
  const float* s2b = (const float*)d_in[13];
  const float* W3  = (const float*)d_in[14];
  const float* as3 = (const float*)d_in[15];
  const float* ad3 = (const float*)d_in[16];
  const float* b3  = (const float*)d_in[17];
  const float* s3W = (const float*)d_in[18];
  const float* s3b = (const float*)d_in[19];
  float* outF = (float*)d_out;

  // workspace carve-up (floats)
  float* ws   = (float*)d_ws;
  float* bufA = ws; ws += (size_t)N_NODES * 256;   // layer1 output
  float* bufB = ws; ws += (size_t)N_NODES * 256;   // layer2 output
  float* lin  = ws; ws += (size_t)N_NODES * 768;   // per-layer X@W
  float* agg3 = ws; ws += (size_t)N_NODES * 768;   // layer3 aggregation
  float* ls   = ws; ws += (size_t)N_NODES * 6;
  float* ld   = ws; ws += (size_t)N_NODES * 6;
  float* mbuf = ws; ws += (size_t)N_NODES * 6;
  float* dnm  = ws; ws += (size_t)N_NODES * 6;
  float* ew   = ws; ws += (size_t)ETOT    * 6;

  auto gemm = [&](const float* A, const float* B, float* C, int K, int Nc) {
    int tiles = (N_NODES / 16) * (Nc / 64);
    gemm_wmma_f32<<<ceil_div(tiles, 8), 256, 0, stream>>>(A, B, C, N_NODES, K, Nc);
  };
  auto attention = [&](const float* linp, float* aggp,
                       const float* a_s, const float* a_d, int H, int Cc) {
    int NH = N_NODES * H;
    int EH = ETOT * H;
    node_logits<<<ceil_div(NH, 256), 256, 0, stream>>>(linp, a_s, a_d, ls, ld, N_NODES, H, Cc);
    init_md<<<ceil_div(NH, 256), 256, 0, stream>>>(mbuf, dnm, NH);
    edge_max<<<ceil_div(EH, 256), 256, 0, stream>>>(ei, ls, ld, mbuf, H);
    edge_expsum<<<ceil_div(EH, 256), 256, 0, stream>>>(ei, ls, ld, mbuf, dnm, ew, H);
    edge_scatter<<<ETOT, 256, 0, stream>>>(ei, linp, ew, dnm, aggp, H, Cc);
  };

  // ---- layer 1: D_IN=128 -> 4x64 concat ----
  gemm(x, W1, lin, 128, 256);
  gemm(x, s1W, bufA, 128, 256);                 // skip pre-loads the aggregator
  attention(lin, bufA, as1, ad1, 4, 64);
  finalize_concat<<<ceil_div(N_NODES * 256, 256), 256, 0, stream>>>(bufA, b1, s1b, bufA, N_NODES, 256);

  // ---- layer 2: 256 -> 4x64 concat ----
  gemm(bufA, W2, lin, 256, 256);
  gemm(bufA, s2W, bufB, 256, 256);
  attention(lin, bufB, as2, ad2, 4, 64);
  finalize_concat<<<ceil_div(N_NODES * 256, 256), 256, 0, stream>>>(bufB, b2, s2b, bufB, N_NODES, 256);

  // ---- layer 3: 256 -> mean over 6 heads of 128 ----
  gemm(bufB, W3, lin, 256, 768);
  gemm(bufB, s3W, outF, 256, 128);              // skip lands directly in d_out
  zero_buf<<<ceil_div(N_NODES * 768, 256), 256, 0, stream>>>(agg3, (size_t)N_NODES * 768);
  attention(lin, agg3, as3, ad3, 6, 128);
  finalize_mean<<<ceil_div(N_NODES * 128, 256), 256, 0, stream>>>(agg3, b3, s3b, outF, N_NODES);
}